// EmptyNet_76647986364466
// MI455X (gfx1250) — compile-verified
//
#include <hip/hip_runtime.h>
#include <math.h>

// Problem constants (from reference): x (8,3,512,512) f32, STRIDE=8, M=0.1
#define BB    8
#define HH    512
#define WW    512
#define SS    8
#define GH    64
#define GW    64
#define NSP   (GH * GW)          // 4096 superpixels per batch
#define CF    5                  // 3 image channels + 2 coords
#define CSCL  0.0125f            // M / STRIDE = 0.1 / 8
#define HW    ((size_t)HH * WW)

typedef __attribute__((ext_vector_type(2))) float v2f;
typedef __attribute__((ext_vector_type(8))) float v8f;

// ---------------------------------------------------------------------------
// Assign: for every pixel, logit_m = 2*(f . c_m) - |c_m|^2 over the 3x3
// neighborhood of superpixel centers (|f|^2 cancels in softmax), softmax over
// the 9.  f.c is V_WMMA_F32_16X16X4_F32: A = centers (16 rows, 9 used,
// K = Cf padded to 8 -> 2 WMMAs), B = 16 pixels.  One wave per 8x8 cell,
// two tiles per iteration so that after 9 lane^16 shuffles ALL 32 lanes own
// a full 9-logit vector and do the softmax.  Fully branch-free hot loop:
// channel selection and coord injection are v_cndmask, not exec masking.
// ---------------------------------------------------------------------------
__global__ __launch_bounds__(256) void slic_assign(const float* __restrict__ x,
                                                   const float* __restrict__ centers,
                                                   float* __restrict__ sims) {
    const int lane  = threadIdx.x & 31;
    const int wave  = threadIdx.x >> 5;
    const int gcell = blockIdx.x * 8 + wave;            // [0, BB*NSP)
    const int b     = gcell / NSP;
    const int cell  = gcell - b * NSP;
    const int gh    = cell / GW;
    const int gw    = cell - gh * GW;

    // per-wave LDS: [0..44] cen[m][ch], [45..53] |c_m|^2
    __shared__ float lds[8][56];
    float* cen = &lds[wave][0];
    float* cnm = &lds[wave][45];

    for (int idx = lane; idx < 45; idx += 32) {
        const int m  = idx / CF, ch = idx - m * CF;
        const int nh = gh + (m / 3 - 1);
        const int nw = gw + (m % 3 - 1);
        const int ph = min(max(nh, 0), GH - 1);
        const int pw = min(max(nw, 0), GW - 1);
        cen[idx] = centers[((size_t)b * NSP + (size_t)(ph * GW + pw)) * CF + ch];
    }
    __syncthreads();
    if (lane < 9) {
        float s = 0.0f;
#pragma unroll
        for (int ch = 0; ch < CF; ++ch) { const float v = cen[lane * CF + ch]; s += v * v; }
        cnm[lane] = s;
    }
    __syncthreads();

    // Per-lane bias[m] = valid ? -|c_m|^2 : -INF  (register array, no LDS in loop)
    float bias[9];
#pragma unroll
    for (int m = 0; m < 9; ++m) {
        const int nh = gh + (m / 3 - 1);
        const int nw = gw + (m % 3 - 1);
        const bool ok = (nh >= 0) & (nh < GH) & (nw >= 0) & (nw < GW);
        bias[m] = ok ? -cnm[m] : -INFINITY;
    }

    // A fragments (loop invariant, branch-free). 16x4 f32 A layout:
    // lanes 0-15: VGPR0=K0, VGPR1=K1 ; lanes 16-31: VGPR0=K2, VGPR1=K3 (M = lane%16)
    const int  mrow = lane & 15;
    const bool hi   = lane >= 16;
    const bool mok  = mrow < 9;
    const int  mr   = min(mrow, 8);                    // clamp for safe LDS read
    const float cA0 = cen[mr * CF + (hi ? 2 : 0)];
    const float cA1 = cen[mr * CF + (hi ? 3 : 1)];
    const float cA4 = cen[mr * CF + 4];
    v2f a0, a1;
    a0.x = mok ? cA0 : 0.0f;
    a0.y = mok ? cA1 : 0.0f;
    a1.x = (mok && !hi) ? cA4 : 0.0f;
    a1.y = 0.0f;

    const size_t xb    = (size_t)b * 3 * HW;
    const size_t obase = (size_t)b * 9 * HW;
    const size_t chA   = hi ? 2 * HW : 0;              // channel offset for first load

#pragma unroll
    for (int tt = 0; tt < 2; ++tt) {
        v8f acc[2];
#pragma unroll
        for (int u = 0; u < 2; ++u) {
            const int p = (2 * tt + u) * 16 + mrow;     // pixel 0..63 inside cell
            const int h = gh * SS + (p >> 3);
            const int w = gw * SS + (p & 7);
            const size_t pix = (size_t)h * WW + w;

            // Branch-free B build (4x16 layout: VGPR0 = K0|K2, VGPR1 = K1|K3)
            const float vA = x[xb + chA + pix];         // lo: ch0, hi: ch2
            const float vB = x[xb + HW  + pix];         // ch1 (all lanes; cache hit for hi)
            v2f b0, b1;
            b0.x = vA;
            b0.y = hi ? (float)h * CSCL : vB;           // hi: K3 = y-coord, lo: K1 = ch1
            b1.x = hi ? 0.0f : (float)w * CSCL;         // lo: K4 = x-coord
            b1.y = 0.0f;

            v8f c = {};
            c = __builtin_amdgcn_wmma_f32_16x16x4_f32(false, a0, false, b0,
                                                      (short)0, c, false, false);
            c = __builtin_amdgcn_wmma_f32_16x16x4_f32(false, a1, false, b1,
                                                      (short)0, c, false, false);
            acc[u] = c;
        }

        // Redistribute so every lane owns a full pixel:
        //   lane p      (<16): tile (2tt+0) pixel p : m0..7 = acc[0][m], m8 from lane p+16
        //   lane p+16        : tile (2tt+1) pixel p : m0..7 from lane p's acc[1][m], m8 = own acc[1][0]
        float xm[8];
#pragma unroll
        for (int m = 0; m < 8; ++m) xm[m] = __shfl(acc[1][m], lane ^ 16, 32);
        const float x8 = __shfl(acc[0][0], lane ^ 16, 32);

        float lg[9];
#pragma unroll
        for (int m = 0; m < 8; ++m) {
            const float dot = hi ? xm[m] : acc[0][m];
            lg[m] = fmaf(dot, 2.0f, bias[m]);
        }
        {
            const float dot8 = hi ? acc[1][0] : x8;
            lg[8] = fmaf(dot8, 2.0f, bias[8]);
        }

        // This lane's pixel coordinates
        const int t = 2 * tt + (hi ? 1 : 0);
        const int p = t * 16 + mrow;
        const int h = gh * SS + (p >> 3);
        const int w = gw * SS + (p & 7);

        float mx = lg[0];
#pragma unroll
        for (int m = 1; m < 9; ++m) mx = fmaxf(mx, lg[m]);
        float e[9], ssum = 0.0f;
#pragma unroll
        for (int m = 0; m < 9; ++m) { e[m] = __expf(lg[m] - mx); ssum += e[m]; }
        const float inv = 1.0f / ssum;
#pragma unroll
        for (int m = 0; m < 9; ++m)
            sims[obase + (size_t)m * HW + (size_t)h * WW + w] = e[m] * inv;
    }
}

// ---------------------------------------------------------------------------
// Update (gather form, deterministic, no atomics): cell g collects, from the
// pixels of each valid neighbor cell c = g+(dh,dw), their weight toward g,
// which is sims channel k=(1-dh)*3+(1-dw).  64 threads = 64 pixels per source
// cell; 6-float LDS tree reduction; centers[g] = num / max(den,1e-8).
// init=1 reproduces the reference's initial per-cell mean (own cell, w=1).
// ---------------------------------------------------------------------------
__global__ __launch_bounds__(64) void slic_update(const float* __restrict__ x,
                                                  const float* __restrict__ sims,
                                                  float* __restrict__ centers,
                                                  int init) {
    const int t     = threadIdx.x;                     // pixel offset 0..63
    const int gcell = blockIdx.x;
    const int b     = gcell / NSP;
    const int cell  = gcell - b * NSP;
    const int gh    = cell / GW;
    const int gw    = cell - gh * GW;

    float num0 = 0.f, num1 = 0.f, num2 = 0.f, num3 = 0.f, num4 = 0.f, den = 0.f;
    const size_t xb = (size_t)b * 3 * HW;
    const int mlo = init ? 4 : 0;
    const int mhi = init ? 5 : 9;

    for (int m9 = mlo; m9 < mhi; ++m9) {
        const int dh = m9 / 3 - 1, dw = m9 % 3 - 1;
        const int sh = gh + dh, sw = gw + dw;
        if (sh < 0 || sh >= GH || sw < 0 || sw >= GW) continue;
        const int h = sh * SS + (t >> 3);
        const int w = sw * SS + (t & 7);
        const size_t pix = (size_t)h * WW + w;
        float wgt = 1.0f;
        if (!init) {
            const int k = (1 - dh) * 3 + (1 - dw);     // neighbor slot pointing back to g
            wgt = sims[((size_t)b * 9 + k) * HW + pix];
        }
        num0 += x[xb + 0 * HW + pix] * wgt;
        num1 += x[xb + 1 * HW + pix] * wgt;
        num2 += x[xb + 2 * HW + pix] * wgt;
        num3 += (float)h * CSCL * wgt;
        num4 += (float)w * CSCL * wgt;
        den  += wgt;
    }

    __shared__ float red[6][64];
    red[0][t] = num0; red[1][t] = num1; red[2][t] = num2;
    red[3][t] = num3; red[4][t] = num4; red[5][t] = den;
    __syncthreads();
#pragma unroll
    for (int s = 32; s > 0; s >>= 1) {
        if (t < s) {
#pragma unroll
            for (int j = 0; j < 6; ++j) red[j][t] += red[j][t + s];
        }
        __syncthreads();
    }
    if (t < CF) {
        const float d = fmaxf(red[5][0], 1e-8f);
        centers[((size_t)b * NSP + cell) * CF + t] = red[t][0] / d;
    }
}

// ---------------------------------------------------------------------------
// d_in[0] : x (8,3,512,512) f32
// d_out   : sims (8,9,512,512) f32 — also used as the inter-iteration sims
//           buffer (fully overwritten by every assign pass).
// d_ws    : centers, BB*NSP*CF floats = 655,360 bytes.
// ---------------------------------------------------------------------------
extern "C" void kernel_launch(void* const* d_in, const int* in_sizes, int n_in,
                              void* d_out, int out_size, void* d_ws, size_t ws_size,
                              hipStream_t stream) {
    const float* x   = (const float*)d_in[0];
    float* sims      = (float*)d_out;
    float* centers   = (float*)d_ws;

    const int assign_blocks = (BB * NSP) / 8;          // 8 waves (cells) per block
    const int update_blocks = BB * NSP;

    // initial centers = per-cell mean of f
    slic_update<<<update_blocks, 64, 0, stream>>>(x, sims, centers, 1);

    for (int it = 0; it < 5; ++it) {
        slic_assign<<<assign_blocks, 256, 0, stream>>>(x, centers, sims);
        slic_update<<<update_blocks, 64, 0, stream>>>(x, sims, centers, 0);
    }
    // final assign -> d_out
    slic_assign<<<assign_blocks, 256, 0, stream>>>(x, centers, sims);
}